// myHeteroConv_34505767256326
// MI455X (gfx1250) — compile-verified
//
#include <hip/hip_runtime.h>
#include <math.h>

typedef __attribute__((ext_vector_type(2))) float v2f;
typedef __attribute__((ext_vector_type(8))) float v8f;

#define FD 128   // feature dim D
#define TD 384   // 3*D
#define BM 64
#define BN 64
#define BK 32
#define LDS_PITCH 34  // padded row pitch (even -> 8B-aligned v2f reads)

// -------------------------------------------------------------------------
// zero helper (grid-stride)
// -------------------------------------------------------------------------
__global__ void zero_kernel(float* __restrict__ p, size_t n) {
  size_t i = (size_t)blockIdx.x * blockDim.x + threadIdx.x;
  size_t stride = (size_t)gridDim.x * blockDim.x;
  for (; i < n; i += stride) p[i] = 0.0f;
}

// -------------------------------------------------------------------------
// Generic fp32 GEMM via V_WMMA_F32_16X16X4_F32.
// C[M,N] = A[M,K] @ B[K,N], row-major. N % 64 == 0, K % 32 == 0 (true here).
// mode 0: C = acc
// mode 1: C = extra[row,col] * sigmoid(acc + bias[col])   (gate fuse, ldExtra==N)
// mode 2: C = (mask[row] != 0) ? 0 : acc + bias[col]      (final aggr)
// Block: 128 threads = 4 waves; block tile 64x64; wave w does rows [16w,16w+16).
// -------------------------------------------------------------------------
__global__ __launch_bounds__(128)
void wmma_gemm_f32(const float* __restrict__ A, const float* __restrict__ B,
                   float* __restrict__ C, int M, int N, int K, int mode,
                   const float* __restrict__ bias, const float* __restrict__ extra) {
  __shared__ float As[BM][LDS_PITCH];    // As[m][k]
  __shared__ float BsT[BN][LDS_PITCH];   // BsT[n][k]  (transposed for contiguous K)

  const int tid  = threadIdx.x;
  const int wave = tid >> 5;
  const int lane = tid & 31;
  const int half = lane >> 4;   // 0: K pair {0,1}; 1: K pair {2,3}
  const int l    = lane & 15;

  const int m0 = blockIdx.x * BM;
  const int n0 = blockIdx.y * BN;

  v8f acc[4] = {};   // four 16x16 N-tiles

  for (int k0 = 0; k0 < K; k0 += BK) {
    // ---- stage A tile (64x32), row-guarded ----
    for (int s = tid; s < BM * (BK / 4); s += 128) {
      const int r = s >> 3, c = s & 7;
      const int gr = m0 + r;
      float4 v = make_float4(0.f, 0.f, 0.f, 0.f);
      if (gr < M) v = *(const float4*)(A + (size_t)gr * K + k0 + c * 4);
      As[r][c * 4 + 0] = v.x; As[r][c * 4 + 1] = v.y;
      As[r][c * 4 + 2] = v.z; As[r][c * 4 + 3] = v.w;
    }
    // ---- stage B tile (32x64) transposed into BsT[n][k] ----
    for (int s = tid; s < BK * (BN / 4); s += 128) {
      const int r = s >> 4, c = s & 15;  // r = k offset, c*4 = n offset
      const float4 v = *(const float4*)(B + (size_t)(k0 + r) * N + n0 + c * 4);
      BsT[c * 4 + 0][r] = v.x; BsT[c * 4 + 1][r] = v.y;
      BsT[c * 4 + 2][r] = v.z; BsT[c * 4 + 3][r] = v.w;
    }
    __syncthreads();

#pragma unroll
    for (int kk = 0; kk < BK; kk += 4) {
      const int kc = kk + 2 * half;
      // A frag 16x4: lane l row m0+16w+l, K = kc, kc+1 (contiguous in LDS)
      const v2f a = *(const v2f*)&As[wave * 16 + l][kc];
#pragma unroll
      for (int nt = 0; nt < 4; nt++) {
        const v2f b = *(const v2f*)&BsT[nt * 16 + l][kc];
        acc[nt] = __builtin_amdgcn_wmma_f32_16x16x4_f32(
            false, a, false, b, (short)0, acc[nt], false, false);
      }
    }
    __syncthreads();
  }

  // ---- epilogue: VGPR r holds M = r (lanes 0-15) / r+8 (lanes 16-31) ----
#pragma unroll
  for (int nt = 0; nt < 4; nt++) {
#pragma unroll
    for (int r = 0; r < 8; r++) {
      const int row = m0 + wave * 16 + r + half * 8;
      const int col = n0 + nt * 16 + l;
      if (row < M) {
        float v = acc[nt][r];
        if (mode == 1) {
          const float g = 1.0f / (1.0f + __expf(-(v + bias[col])));
          v = extra[(size_t)row * N + col] * g;
        } else if (mode == 2) {
          v += bias[col];
          if (extra[row] != 0.0f) v = 0.0f;
        }
        C[(size_t)row * N + col] = v;
      }
    }
  }
}

// -------------------------------------------------------------------------
// Scatter: one wave per edge. Gather 512B transformed row (float4/lane),
// atomic-add into total[:,0:128] (sum), atomic-fmax into total[:,256:384].
// Accumulator (153.6 MB) fits the 192 MB L2 -> atomics resolve on-chip.
// -------------------------------------------------------------------------
__global__ __launch_bounds__(256)
void scatter_kernel(const float* __restrict__ h, const int* __restrict__ src,
                    const int* __restrict__ dst, int E,
                    float* __restrict__ total, float* __restrict__ cnt) {
  const int wid  = (int)(((size_t)blockIdx.x * blockDim.x + threadIdx.x) >> 5);
  const int lane = threadIdx.x & 31;
  if (wid >= E) return;
  __builtin_prefetch(src + wid + 2048, 0, 0);  // global_prefetch_b8 ahead on index stream
  const int s = src[wid];
  const int d = dst[wid];
  const float4 v = *(const float4*)(h + (size_t)s * FD + lane * 4);
  float* trow = total + (size_t)d * TD + lane * 4;
  atomicAdd(trow + 0, v.x);
  atomicAdd(trow + 1, v.y);
  atomicAdd(trow + 2, v.z);
  atomicAdd(trow + 3, v.w);
  float* mrow = trow + 2 * FD;
  __hip_atomic_fetch_max(mrow + 0, v.x, __ATOMIC_RELAXED, __HIP_MEMORY_SCOPE_AGENT);
  __hip_atomic_fetch_max(mrow + 1, v.y, __ATOMIC_RELAXED, __HIP_MEMORY_SCOPE_AGENT);
  __hip_atomic_fetch_max(mrow + 2, v.z, __ATOMIC_RELAXED, __HIP_MEMORY_SCOPE_AGENT);
  __hip_atomic_fetch_max(mrow + 3, v.w, __ATOMIC_RELAXED, __HIP_MEMORY_SCOPE_AGENT);
  if (lane == 0) atomicAdd(cnt + d, 1.0f);
}

// -------------------------------------------------------------------------
// Finalize: one wave per node. mean block = sum / clamp(cnt,1);
// mask[v] = (sum |total_row|) == 0, via wave32 shuffle reduction.
// (max block was init-0 + fmax => already equals max(segment_max, 0).)
// -------------------------------------------------------------------------
__global__ __launch_bounds__(256)
void finalize_kernel(float* __restrict__ total, const float* __restrict__ cnt,
                     float* __restrict__ maskv, int NV) {
  const int wid  = (int)(((size_t)blockIdx.x * blockDim.x + threadIdx.x) >> 5);
  const int lane = threadIdx.x & 31;
  if (wid >= NV) return;
  const float inv = 1.0f / fmaxf(cnt[wid], 1.0f);
  float* row = total + (size_t)wid * TD;
  const float4 s4 = *(const float4*)(row + lane * 4);
  const float4 m4 = *(const float4*)(row + 2 * FD + lane * 4);
  float4 mean;
  mean.x = s4.x * inv; mean.y = s4.y * inv;
  mean.z = s4.z * inv; mean.w = s4.w * inv;
  *(float4*)(row + FD + lane * 4) = mean;
  float ab = fabsf(s4.x) + fabsf(s4.y) + fabsf(s4.z) + fabsf(s4.w)
           + fabsf(mean.x) + fabsf(mean.y) + fabsf(mean.z) + fabsf(mean.w)
           + fabsf(m4.x) + fabsf(m4.y) + fabsf(m4.z) + fabsf(m4.w);
  for (int off = 16; off > 0; off >>= 1) ab += __shfl_xor(ab, off, 32);
  if (lane == 0) maskv[wid] = (ab == 0.0f) ? 1.0f : 0.0f;
}

// -------------------------------------------------------------------------
extern "C" void kernel_launch(void* const* d_in, const int* in_sizes, int n_in,
                              void* d_out, int out_size, void* d_ws, size_t ws_size,
                              hipStream_t stream) {
  const float* x_u    = (const float*)d_in[0];
  const float* x_v    = (const float*)d_in[1];
  const float* W_r1   = (const float*)d_in[2];
  const float* W_r2   = (const float*)d_in[3];
  const float* W_gate = (const float*)d_in[4];
  const float* b_gate = (const float*)d_in[5];
  const float* W_aggr = (const float*)d_in[6];
  const float* b_aggr = (const float*)d_in[7];
  const int*   ei1    = (const int*)d_in[8];   // [2, E1]: row0=src, row1=dst
  const int*   ei2    = (const int*)d_in[9];   // [2, E2]

  const int NU = in_sizes[0] / FD;
  const int NV = in_sizes[1] / FD;
  const int E1 = in_sizes[8] / 2;
  const int E2 = in_sizes[9] / 2;

  // workspace carve-out
  float* h_u   = (float*)d_ws;                  // NU*128
  float* h_v   = h_u + (size_t)NU * FD;         // NV*128
  float* total = h_v + (size_t)NV * FD;         // NV*384 (sum | mean | max)
  float* gated = total + (size_t)NV * TD;       // NV*384
  float* cnt   = gated + (size_t)NV * TD;       // NV
  float* maskv = cnt + NV;                      // NV

  zero_kernel<<<2048, 256, 0, stream>>>(total, (size_t)NV * TD);
  zero_kernel<<<256, 256, 0, stream>>>(cnt, (size_t)NV);

  const dim3 blk(128);
  // per-node relation transforms (hoisted out of the per-edge loop)
  dim3 g1((NU + BM - 1) / BM, FD / BN);
  wmma_gemm_f32<<<g1, blk, 0, stream>>>(x_u, W_r1, h_u, NU, FD, FD, 0, nullptr, nullptr);
  dim3 g2((NV + BM - 1) / BM, FD / BN);
  wmma_gemm_f32<<<g2, blk, 0, stream>>>(x_v, W_r2, h_v, NV, FD, FD, 0, nullptr, nullptr);

  // edge scatter (sum + max + count), one wave per edge
  scatter_kernel<<<(E1 + 7) / 8, 256, 0, stream>>>(h_u, ei1, ei1 + E1, E1, total, cnt);
  scatter_kernel<<<(E2 + 7) / 8, 256, 0, stream>>>(h_v, ei2, ei2 + E2, E2, total, cnt);

  // mean + empty-row mask
  finalize_kernel<<<(NV + 7) / 8, 256, 0, stream>>>(total, cnt, maskv, NV);

  // gated = total * sigmoid(total @ W_gate + b_gate)
  dim3 g3((NV + BM - 1) / BM, TD / BN);
  wmma_gemm_f32<<<g3, blk, 0, stream>>>(total, W_gate, gated, NV, TD, TD, 1, b_gate, total);

  // out = mask ? 0 : gated @ W_aggr + b_aggr
  dim3 g4((NV + BM - 1) / BM, FD / BN);
  wmma_gemm_f32<<<g4, blk, 0, stream>>>(gated, W_aggr, (float*)d_out, NV, FD, TD, 2, b_aggr, maskv);
}